// PositionEmbeddingSine3d_21320217657410
// MI455X (gfx1250) — compile-verified
//
#include <hip/hip_runtime.h>
#include <hip/hip_bf16.h>
#include <stdint.h>

// ---------------------------------------------------------------------------
// Stable batched scatter for MI455X (gfx1250, wave32).
// Memory-bound: ~1.1 GB total traffic -> ~47us at 23.3 TB/s.
// CDNA5 path: async global<->LDS copies (ASYNCcnt) for 256B row moves.
// Builtin signature (probe-confirmed by round-1 diagnostic):
//   __builtin_amdgcn_global_load_async_to_lds_b64(
//       __attribute__((address_space(1))) v2i* gaddr,
//       __attribute__((address_space(3))) v2i* lds, imm int off, imm int cpol)
// ---------------------------------------------------------------------------

#define MAXB          16     // max batch size supported (actual bs = 8)
#define CHUNK         4096   // rows per chunk (=> 489 chunks for N=2M, <=512 for scan)
#define K3_THREADS    256    // 8 wave32 waves
#define NWAVES        8
#define SUBCHUNKS     128    // CHUNK / 32
#define SUBS_PER_WAVE 16     // SUBCHUNKS / NWAVES

#if defined(__gfx1250__) && \
    __has_builtin(__builtin_amdgcn_global_load_async_to_lds_b64) && \
    __has_builtin(__builtin_amdgcn_global_store_async_from_lds_b64)
#define HAVE_ASYNC_LDS 1
#else
#define HAVE_ASYNC_LDS 0
#endif

typedef int v2i __attribute__((vector_size(2 * sizeof(int))));   // int2 vector
typedef __attribute__((address_space(1))) v2i gv2i;              // global int2
typedef __attribute__((address_space(3))) v2i lv2i;              // LDS int2

#if HAVE_ASYNC_LDS
__device__ __forceinline__ void wait_asynccnt0() {
#if __has_builtin(__builtin_amdgcn_s_wait_asynccnt)
  __builtin_amdgcn_s_wait_asynccnt(0);
#else
  asm volatile("s_wait_asynccnt 0" ::: "memory");
#endif
}
#endif

// ---------------- Pass 1: per-chunk histogram of batch ids ------------------
__global__ void k1_hist(const int* __restrict__ inds, int* __restrict__ blockHist, int N) {
  __shared__ int lh[MAXB];
  const int t = threadIdx.x;
  if (t < MAXB) lh[t] = 0;
  __syncthreads();
  const int base = blockIdx.x * CHUNK;
  const int end  = min(base + CHUNK, N);
  for (int r = base + t; r < end; r += blockDim.x) {
    int b = inds[(size_t)r * 4];  // column 0 of (N,4)
    if ((unsigned)b < MAXB) atomicAdd(&lh[b], 1);
  }
  __syncthreads();
  if (t < MAXB) blockHist[blockIdx.x * MAXB + t] = lh[t];
}

// ---------------- Pass 2: exclusive scan over chunks, per batch -------------
// grid = MAXB blocks, 512 threads. Requires nblocks <= 512 (489 for N=2M).
__global__ void k2_scan(int* __restrict__ blockHist, int* __restrict__ counts, int nblocks) {
  __shared__ int sd[512];
  const int b = blockIdx.x;
  const int t = threadIdx.x;
  int v = (t < nblocks) ? blockHist[(size_t)t * MAXB + b] : 0;
  sd[t] = v;
  __syncthreads();
  for (int off = 1; off < 512; off <<= 1) {
    int x = (t >= off) ? sd[t - off] : 0;
    __syncthreads();
    sd[t] += x;
    __syncthreads();
  }
  if (t < nblocks) blockHist[(size_t)t * MAXB + b] = sd[t] - v;  // exclusive
  if (t == 511)   counts[b] = sd[511];                           // batch total
}

// ---------------- Pass 3: stable rank + scatter rows ------------------------
__global__ __launch_bounds__(K3_THREADS) void k3_scatter(
    const int* __restrict__ inds, const float* __restrict__ feats,
    const int* __restrict__ blockHist, float* __restrict__ out,
    int N, int C, const int* __restrict__ bs_p, const int* __restrict__ ml_p) {
  __shared__ int      subhist[SUBCHUNKS * MAXB];
  __shared__ unsigned posb[CHUNK];   // packed (b<<24)|pos, 0xFFFFFFFF = skip
  __shared__ int      boff[MAXB];
#if HAVE_ASYNC_LDS
  __shared__ float    stage[NWAVES * 8 * 64];  // 8 rows x 64 f32 per wave
#endif
  const int      t      = threadIdx.x;
  const int      w      = t >> 5;        // wave id (wave32)
  const int      lane   = t & 31;
  const unsigned laneLt = (1u << lane) - 1u;
  const int      bs     = *bs_p;
  const int      ml     = *ml_p;
  const int      base   = blockIdx.x * CHUNK;

  if (t < MAXB) boff[t] = blockHist[blockIdx.x * MAXB + t];

  // Phase A: sub-chunk (32-row) histograms + per-lane stable ranks via ballot
  int bArr[SUBS_PER_WAVE];
  int rkArr[SUBS_PER_WAVE];
  for (int k = 0; k < SUBS_PER_WAVE; ++k) {
    const int s   = w * SUBS_PER_WAVE + k;       // sub-chunk in row order
    const int row = base + s * 32 + lane;
    const int b   = (row < N) ? inds[(size_t)row * 4] : -1;
    int rk = 0;
    for (int v = 0; v < bs; ++v) {
      unsigned m = (unsigned)__ballot(b == v);   // wave32: low 32 bits
      if (b == v)    rk = __popc(m & laneLt);
      if (lane == v) subhist[s * MAXB + v] = __popc(m);
    }
    bArr[k]  = b;
    rkArr[k] = rk;
  }
  __syncthreads();

  // Phase B: exclusive scan of sub-chunk histograms (serial per batch, 128 steps)
  if (t < bs) {
    int run = 0;
    for (int s = 0; s < SUBCHUNKS; ++s) {
      int x = subhist[s * MAXB + t];
      subhist[s * MAXB + t] = run;
      run += x;
    }
  }
  __syncthreads();

  // Phase C: final positions, packed into LDS
  for (int k = 0; k < SUBS_PER_WAVE; ++k) {
    const int s = w * SUBS_PER_WAVE + k;
    const int b = bArr[k];
    unsigned pack = 0xFFFFFFFFu;
    if (b >= 0 && b < bs) {
      int pos = boff[b] + subhist[s * MAXB + b] + rkArr[k];
      if (pos < ml) pack = ((unsigned)b << 24) | (unsigned)pos;
    }
    posb[s * 32 + lane] = pack;
  }
  __syncthreads();

  // Phase D: coalesced 256B row copies (wave-cooperative)
  const int rowsBegin = w * (CHUNK / NWAVES);    // 512 contiguous rows per wave
  if (C == 64) {
    for (int j0 = 0; j0 < CHUNK / NWAVES; j0 += 32) {
      const unsigned packv = posb[rowsBegin + j0 + lane];
#if HAVE_ASYNC_LDS
      // Pipeline: 8 async loads -> wait -> 8 async stores -> wait
      for (int k0 = 0; k0 < 32; k0 += 8) {
        for (int k = k0; k < k0 + 8; ++k) {
          const unsigned pk = __shfl(packv, k, 32);
          if (pk == 0xFFFFFFFFu) continue;                 // wave-uniform branch
          const int row = base + rowsBegin + j0 + k;
          const float* src = feats + (size_t)row * 64 + lane * 2;
          float* dstStage  = &stage[(w * 8 + (k - k0)) * 64 + lane * 2];
          __builtin_amdgcn_global_load_async_to_lds_b64(
              (gv2i*)(uintptr_t)src,
              (lv2i*)(uint32_t)(uintptr_t)dstStage, 0, 0);
        }
        wait_asynccnt0();
        for (int k = k0; k < k0 + 8; ++k) {
          const unsigned pk = __shfl(packv, k, 32);
          if (pk == 0xFFFFFFFFu) continue;
          float* dst = out + ((size_t)(pk >> 24) * (size_t)ml + (pk & 0xFFFFFFu)) * 64
                           + lane * 2;
          const float* srcStage = &stage[(w * 8 + (k - k0)) * 64 + lane * 2];
          __builtin_amdgcn_global_store_async_from_lds_b64(
              (gv2i*)(uintptr_t)dst,
              (lv2i*)(uint32_t)(uintptr_t)srcStage, 0, 0);
        }
        wait_asynccnt0();   // LDS staging reused next group
      }
#else
      for (int k = 0; k < 32; ++k) {
        const unsigned pk = __shfl(packv, k, 32);
        if (pk == 0xFFFFFFFFu) continue;
        const int row = base + rowsBegin + j0 + k;
        const float2* src = (const float2*)(feats + (size_t)row * 64);
        float2* dst = (float2*)(out + ((size_t)(pk >> 24) * (size_t)ml + (pk & 0xFFFFFFu)) * 64);
        dst[lane] = src[lane];   // 32 lanes x 8B = 256B coalesced
      }
#endif
    }
  } else {
    // generic C fallback
    for (int j = 0; j < CHUNK / NWAVES; ++j) {
      const int lr = rowsBegin + j;
      const unsigned pk = posb[lr];
      if (pk == 0xFFFFFFFFu) continue;
      const int row = base + lr;
      const float* src = feats + (size_t)row * C;
      float* dst = out + ((size_t)(pk >> 24) * (size_t)ml + (pk & 0xFFFFFFu)) * C;
      for (int c = lane; c < C; c += 32) dst[c] = src[c];
    }
  }
}

// ---------------- Pass 4: zero only the tail rows ---------------------------
__global__ void k4_zero_tail(const int* __restrict__ counts, float* __restrict__ out,
                             int C, long rowsTotal, const int* __restrict__ ml_p) {
  long r = (long)blockIdx.x * blockDim.x + threadIdx.x;
  const long stride = (long)gridDim.x * blockDim.x;
  const int ml = *ml_p;
  for (; r < rowsTotal; r += stride) {
    const int b   = (int)(r / ml);
    const int pos = (int)(r - (long)b * ml);
    if (pos >= counts[b]) {
      float* dst = out + (size_t)r * C;
      if ((C & 3) == 0) {
        float4 z = {0.f, 0.f, 0.f, 0.f};
        float4* d4 = (float4*)dst;
        for (int j = 0; j < (C >> 2); ++j) d4[j] = z;
      } else {
        for (int j = 0; j < C; ++j) dst[j] = 0.f;
      }
    }
  }
}

// ---------------------------------------------------------------------------
extern "C" void kernel_launch(void* const* d_in, const int* in_sizes, int n_in,
                              void* d_out, int out_size, void* d_ws, size_t ws_size,
                              hipStream_t stream) {
  const int*   inds  = (const int*)d_in[0];
  const float* feats = (const float*)d_in[1];
  const int*   bs_p  = (const int*)d_in[2];
  const int*   ml_p  = (const int*)d_in[3];
  float*       out   = (float*)d_out;

  const int N = in_sizes[0] / 4;                 // inds is (N,4)
  const int C = (N > 0) ? (in_sizes[1] / N) : 1; // features is (N,C)
  const int nblocks = (N + CHUNK - 1) / CHUNK;   // 489 for N=2M (<=512 required by k2)

  int* blockHist = (int*)d_ws;                         // nblocks*MAXB ints
  int* counts    = blockHist + (size_t)nblocks * MAXB; // MAXB ints

  k1_hist<<<nblocks, 256, 0, stream>>>(inds, blockHist, N);
  k2_scan<<<MAXB, 512, 0, stream>>>(blockHist, counts, nblocks);
  k3_scatter<<<nblocks, K3_THREADS, 0, stream>>>(inds, feats, blockHist, out, N, C, bs_p, ml_p);

  const long rowsTotal = (long)out_size / (C > 0 ? C : 1);
  long zb = (rowsTotal + 255) / 256;
  if (zb > 16384) zb = 16384;
  if (zb < 1) zb = 1;
  k4_zero_tail<<<(int)zb, 256, 0, stream>>>(counts, out, C, rowsTotal, ml_p);
}